// QuantumAutoencoder_65481071399199
// MI455X (gfx1250) — compile-verified
//
#include <hip/hip_runtime.h>

// QuantumAutoencoder expectation-value map:
//   out = cos(in), with out[:,1] = cos(in[:,0]) * cos(in[:,1])
// Shape: (2097152 rows, 16 cols) f32 -> f32. Purely elementwise, streaming.
//
// Roofline (MI455X): 256 MiB total traffic @ 23.3 TB/s => ~11.5 us floor.
// 33.5M cosines via hardware v_cos_f32 (~1-2 us of trans-VALU) keeps us
// memory-bound, so the kernel is built around wide (b128) non-temporal
// streaming loads/stores; data (256 MB) exceeds the 192 MB L2 and is
// touched once, so NT hints avoid L2 thrash.

typedef float v4f __attribute__((ext_vector_type(4)));

__global__ __launch_bounds__(256) void qae_cosz_kernel(
    const v4f* __restrict__ in, v4f* __restrict__ out, int n4) {
  int i = blockIdx.x * blockDim.x + threadIdx.x;
  if (i >= n4) return;

  // 512 contiguous bytes per wave32 per load: global_load_b128 th:NT
  v4f a = __builtin_nontemporal_load(in + i);

  v4f c;
  c.x = __cosf(a.x);  // v_mul_f32 (1/2pi) + v_cos_f32
  c.y = __cosf(a.y);
  c.z = __cosf(a.z);
  c.w = __cosf(a.w);

  // Row stride = 16 floats = 4 float4s. The float4 at the row start
  // (i % 4 == 0) holds columns 0..3 of a row: apply the CX(q0->q1)
  // parity fixup <Z1> = cos(t0)*cos(t1) locally (branchless predication,
  // 8 of 32 lanes active per wave).
  if ((i & 3) == 0) c.y *= c.x;

  __builtin_nontemporal_store(c, out + i);  // global_store_b128 th:NT
}

extern "C" void kernel_launch(void* const* d_in, const int* in_sizes, int n_in,
                              void* d_out, int out_size, void* d_ws, size_t ws_size,
                              hipStream_t stream) {
  (void)n_in; (void)d_ws; (void)ws_size; (void)out_size;

  const float* in = (const float*)d_in[0];
  float* out = (float*)d_out;

  int n  = in_sizes[0];   // 2097152 * 16 = 33,554,432 floats (multiple of 4)
  int n4 = n >> 2;        // 8,388,608 float4 work-items

  const int threads = 256;                       // 8 wave32s per block
  const int blocks  = (n4 + threads - 1) / threads;  // 32768 blocks

  qae_cosz_kernel<<<blocks, threads, 0, stream>>>(
      (const v4f*)in, (v4f*)out, n4);
}